// SemanticLayer_69329362092575
// MI455X (gfx1250) — compile-verified
//
#include <hip/hip_runtime.h>
#include <hip/hip_bf16.h>

// Cosine-similarity gather + mean:  out = 1 - mean_i( cos(x[i], s[label[i]]) )
// Memory-bound streaming pass; dots computed with V_WMMA_F32_16X16X4_F32
// (exact einsum shape of the reference: [16x4] x [4x16] -> [16x16]).

typedef __attribute__((ext_vector_type(2))) float v2f;
typedef __attribute__((ext_vector_type(8))) float v8f;
typedef __attribute__((ext_vector_type(4))) int   v4i;

#define COS_EPS    1e-8f
#define COS_EPS_SQ 1e-16f
#define NCLASS 10

__global__ __launch_bounds__(256)
void SemanticLayer_cos_kernel(const float* __restrict__ x,
                              const int* __restrict__ label,
                              const float* __restrict__ s,
                              float* __restrict__ partials,
                              long nB) {
    const int lane = threadIdx.x & 31;
    const int n    = lane & 15;   // class column (B/D) / sample row (A)
    const int hi   = lane >> 4;   // lane-half: selects K pair (A) / M+8 (D)

    const int wavesPerBlock = blockDim.x >> 5;
    const long waveId = (long)blockIdx.x * wavesPerBlock + (threadIdx.x >> 5);
    const long nWaves = (long)gridDim.x * wavesPerBlock;

    // ---- Build B = s^T, pre-scaled by 1/max(||s_row||, eps). ----
    // Layout (32-bit B 4x16): VGPR0/1 lane n -> (K=2*hi+0 / 2*hi+1, N=n).
    v2f bmat; bmat.x = 0.0f; bmat.y = 0.0f;
    if (n < NCLASS) {
        bmat.x = s[n * 4 + 2 * hi + 0];
        bmat.y = s[n * 4 + 2 * hi + 1];
    }
    {
        float p  = bmat.x * bmat.x + bmat.y * bmat.y;
        float n2 = p + __shfl_xor(p, 16);            // full ||s_n||^2 (both halves)
        // 1/max(||s||,eps) == rsq(max(||s||^2, eps^2))  -> single v_rsq_f32
        float inv = __builtin_amdgcn_rsqf(fmaxf(n2, COS_EPS_SQ));
        bmat.x *= inv; bmat.y *= inv;
    }

    float acc = 0.0f;
    const long totalTiles = nB >> 5;                 // 32 samples per wave-iteration

    for (long t = waveId; t < totalTiles; t += nWaves) {
        const long base = t << 5;

        // A tiles: lane L holds x[base+L][0..1], lane 16+L holds x[base+L][2..3]
        // -> 8B/lane, 256B contiguous per wave per tile half. Fully coalesced.
        const float* xp = x + ((base + n) << 2) + (hi << 1);
        v2f a1 = *(const v2f*)(xp);                  // samples base    .. base+15
        v2f a2 = *(const v2f*)(xp + 64);             // samples base+16 .. base+31

        // Labels this lane compares against are contiguous runs of 8:
        //   d1 -> label[base + 8*hi .. +7], d2 -> label[base + 16 + 8*hi .. +7]
        // Half-wave lanes share the same 32B -> served from L0, no bpermutes.
        const int* lp = label + base + (hi << 3);
        v4i lA0 = *(const v4i*)(lp);                 // d1, v = 0..3
        v4i lA1 = *(const v4i*)(lp + 4);             // d1, v = 4..7
        v4i lB0 = *(const v4i*)(lp + 16);            // d2, v = 0..3
        v4i lB1 = *(const v4i*)(lp + 20);            // d2, v = 4..7

        // Fold 1/max(||x||,eps) into A (each lane holds half its sample's comps)
        float p1 = a1.x * a1.x + a1.y * a1.y;
        float p2 = a2.x * a2.x + a2.y * a2.y;
        float i1 = __builtin_amdgcn_rsqf(fmaxf(p1 + __shfl_xor(p1, 16), COS_EPS_SQ));
        float i2 = __builtin_amdgcn_rsqf(fmaxf(p2 + __shfl_xor(p2, 16), COS_EPS_SQ));
        a1.x *= i1; a1.y *= i1;
        a2.x *= i2; a2.y *= i2;

        v8f c = {0.f, 0.f, 0.f, 0.f, 0.f, 0.f, 0.f, 0.f};
        // D = A x B : full 16x16 cos matrix (cols >= NCLASS are zero)
        v8f d1 = __builtin_amdgcn_wmma_f32_16x16x4_f32(
            false, a1, false, bmat, (short)0, c, false, false);
        v8f d2 = __builtin_amdgcn_wmma_f32_16x16x4_f32(
            false, a2, false, bmat, (short)0, c, false, false);

        // D cell (M = v + 8*hi, N = n) contributes iff label[sample M] == n.
        #pragma unroll
        for (int v = 0; v < 4; ++v) {
            acc += (lA0[v] == n) ? d1[v]     : 0.0f;
            acc += (lA1[v] == n) ? d1[v + 4] : 0.0f;
            acc += (lB0[v] == n) ? d2[v]     : 0.0f;
            acc += (lB1[v] == n) ? d2[v + 4] : 0.0f;
        }
    }

    // ---- Scalar tail (nB not a multiple of 32; empty for B = 8388608) ----
    {
        const long tailStart = totalTiles << 5;
        for (long i = tailStart + (long)blockIdx.x * blockDim.x + threadIdx.x;
             i < nB; i += (long)gridDim.x * blockDim.x) {
            float x0 = x[i * 4 + 0], x1 = x[i * 4 + 1];
            float x2 = x[i * 4 + 2], x3 = x[i * 4 + 3];
            int   lb = label[i];
            float s0 = s[lb * 4 + 0], s1 = s[lb * 4 + 1];
            float s2 = s[lb * 4 + 2], s3 = s[lb * 4 + 3];
            float dot = x0 * s0 + x1 * s1 + x2 * s2 + x3 * s3;
            float ixn = __builtin_amdgcn_rsqf(
                fmaxf(x0 * x0 + x1 * x1 + x2 * x2 + x3 * x3, COS_EPS_SQ));
            float isn = __builtin_amdgcn_rsqf(
                fmaxf(s0 * s0 + s1 * s1 + s2 * s2 + s3 * s3, COS_EPS_SQ));
            acc += dot * ixn * isn;
        }
    }

    // ---- Deterministic block reduction -> one partial per block ----
    __shared__ float red[256];
    red[threadIdx.x] = acc;
    __syncthreads();
    for (int stride = blockDim.x >> 1; stride > 0; stride >>= 1) {
        if ((int)threadIdx.x < stride) red[threadIdx.x] += red[threadIdx.x + stride];
        __syncthreads();
    }
    if (threadIdx.x == 0) partials[blockIdx.x] = red[0];
}

__global__ __launch_bounds__(256)
void SemanticLayer_finalize_kernel(const float* __restrict__ partials,
                                   int nPartials, float* __restrict__ out,
                                   float invB) {
    __shared__ float red[256];
    float a = 0.0f;
    for (int i = threadIdx.x; i < nPartials; i += blockDim.x) a += partials[i];
    red[threadIdx.x] = a;
    __syncthreads();
    for (int stride = blockDim.x >> 1; stride > 0; stride >>= 1) {
        if ((int)threadIdx.x < stride) red[threadIdx.x] += red[threadIdx.x + stride];
        __syncthreads();
    }
    if (threadIdx.x == 0) out[0] = 1.0f - red[0] * invB;
}

extern "C" void kernel_launch(void* const* d_in, const int* in_sizes, int n_in,
                              void* d_out, int out_size, void* d_ws, size_t ws_size,
                              hipStream_t stream) {
    const float* x     = (const float*)d_in[0];
    const int*   label = (const int*)d_in[1];
    const float* s     = (const float*)d_in[2];
    float* out = (float*)d_out;
    float* partials = (float*)d_ws;

    const long nB = (long)in_sizes[0] / 4;   // x is [B,4] f32

    const int block = 256;
    const int grid  = 1024;                  // 8192 waves; 8.4M/32/8192 = 32 tiles/wave

    SemanticLayer_cos_kernel<<<grid, block, 0, stream>>>(x, label, s, partials, nB);
    SemanticLayer_finalize_kernel<<<1, block, 0, stream>>>(
        partials, grid, out, 1.0f / (float)nB);
}